// LocalConvolution_36172214567834
// MI455X (gfx1250) — compile-verified
//
#include <hip/hip_runtime.h>

typedef float v2f __attribute__((ext_vector_type(2)));
typedef float v8f __attribute__((ext_vector_type(8)));
typedef unsigned int v4u __attribute__((ext_vector_type(4)));
typedef int v8i __attribute__((ext_vector_type(8)));
typedef int v4i __attribute__((ext_vector_type(4)));

#define T_DIM   64
#define C_IN    32
#define H       32
#define W       32
#define C_OUT   64
#define KSZ     5
#define ROWS    28
#define COLS    28
#define KDIM    800                         // C_IN * KSZ * KSZ
#define KC      80                          // K-chunk for B staging
#define NCHUNK  10                          // 800 / 80
#define A_FLOATS (T_DIM * KDIM)             // 51200 floats = 200 KB
#define B_FLOATS (C_OUT * KC)               // 5120 floats  = 20 KB / buffer
#define LDS_BYTES ((A_FLOATS + 2 * B_FLOATS) * 4)   // 245760 B <= 320 KB

extern "C" __global__ void localconv_wmma(const float* __restrict__ in,
                                          const float* __restrict__ wgt,
                                          int* __restrict__ out) {
  extern __shared__ float smem[];
  float* Alds = smem;                  // A[t][k], t in [0,64), k in [0,800)
  float* Blds = smem + A_FLOATS;       // B[buf][o][k_local], o in [0,64), k in [0,80)

  const int rc   = blockIdx.x;         // 0..783, one (r,c) per workgroup
  const int r    = rc / COLS;
  const int c    = rc - r * COLS;
  const int tid  = threadIdx.x;
  const int lane = tid & 31;
  const int wid  = tid >> 5;           // 8 waves

  // ---------------- TDM: gather all patches A[t][k] into LDS -------------
  // 4-D tile: dim0=kw(5,stride 1), dim1=kh(5,stride W=32),
  //           dim2=i(32,stride H*W=1024), dim3=t(64,stride C_IN*H*W=32768)
  if (wid == 0) {
    unsigned long long gaddr =
        (unsigned long long)(const void*)in +
        (unsigned long long)((unsigned)(r * W + c) * 4u);
    unsigned int lds_base = __builtin_amdgcn_groupstaticsize();  // dyn-LDS start

    v4u g0;
    g0[0] = 1u;                                          // count=1, user mode
    g0[1] = lds_base;                                    // lds_addr (bytes)
    g0[2] = (unsigned int)(gaddr & 0xffffffffull);       // global_addr[31:0]
    g0[3] = (unsigned int)((gaddr >> 32) & 0x1ffffffu)   // global_addr[56:32]
          | (2u << 30);                                  // type = 2 ("image")

    v8i g1;
    g1[0] = (int)(2u << 16);            // data_size = 4 bytes
    g1[1] = (int)(5u << 16);            // tensor_dim0 = 5   (>= tile extent)
    g1[2] = (int)(5u << 16);            // tensor_dim1 = 5
    g1[3] = (int)(5u << 16);            // tile_dim0   = 5   (kw)
    g1[4] = (int)(5u | (32u << 16));    // tile_dim1 = 5 (kh), tile_dim2 = 32 (i)
    g1[5] = (int)W;                     // tensor_dim0_stride = 32      (kh step)
    g1[6] = (int)((unsigned)(H * W) << 16);  // tensor_dim1_stride[15:0] = 1024 (i step)
    g1[7] = 0;                          // tensor_dim1_stride[47:16] = 0

    v4i g2;
    g2[0] = 32;                         // tensor_dim2 = 32
    g2[1] = 64;                         // tensor_dim3 = 64
    g2[2] = C_IN * H * W;               // tensor_dim2_stride = 32768 (t step)
    g2[3] = (int)(64u << 16);           // tile_dim3 = 64 (t)

    v4i g3;
    g3[0] = 0; g3[1] = 0; g3[2] = 0; g3[3] = 0;

    v8i g4;                             // extra descriptor group (clang-23 form)
    g4[0] = 0; g4[1] = 0; g4[2] = 0; g4[3] = 0;
    g4[4] = 0; g4[5] = 0; g4[6] = 0; g4[7] = 0;

    __builtin_amdgcn_tensor_load_to_lds(g0, g1, g2, g3, g4, 0);
  }

  // ------------- cooperative B staging (weights, double buffered) --------
  const float* wbase = wgt + (size_t)rc * (C_OUT * KDIM);
  float4 regs[5];

  auto loadB = [&](int ch) {
#pragma unroll
    for (int j = 0; j < 5; ++j) {
      int idx = tid + j * 256;          // 1280 float4's per chunk
      int o   = idx / 20;               // 20 float4 per o-row of 80 floats
      int kq  = idx - o * 20;
      regs[j] = *(const float4*)(wbase + o * KDIM + ch * KC + kq * 4);
    }
  };
  auto storeB = [&](int buf) {
#pragma unroll
    for (int j = 0; j < 5; ++j) {
      int idx = tid + j * 256;
      int o   = idx / 20;
      int kq  = idx - o * 20;
      *(float4*)(Blds + buf * B_FLOATS + o * KC + kq * 4) = regs[j];
    }
  };

  // stage chunk 0 while the TDM streams A
  loadB(0);
  storeB(0);
  if (wid == 0) __builtin_amdgcn_s_wait_tensorcnt(0);
  __syncthreads();

  // ------------------------- WMMA main loop ------------------------------
  // 4x4 grid of 16x16 tiles; wave w: n_tile = w&3, m tiles {2*(w>>2), 2*(w>>2)+1}
  const int nbase  = (wid & 3) * 16;
  const int mbase0 = (wid >> 2) * 32;
  const int mbase1 = mbase0 + 16;
  const int row16  = lane & 15;         // A row / B column within tile
  const int koff   = (lane >> 4) * 2;   // lanes 16-31 hold K+2,K+3

  v8f acc0 = {};
  v8f acc1 = {};

  const float* A0base = Alds + (mbase0 + row16) * KDIM + koff;
  const float* A1base = Alds + (mbase1 + row16) * KDIM + koff;
  const float* Bbase  = Blds + (nbase + row16) * KC + koff;

  for (int ch = 0; ch < NCHUNK; ++ch) {
    if (ch + 1 < NCHUNK) loadB(ch + 1);       // overlap next global fetch

    const float* A0 = A0base + ch * KC;
    const float* A1 = A1base + ch * KC;
    const float* Bb = Bbase + (ch & 1) * B_FLOATS;

#pragma unroll
    for (int s = 0; s < KC; s += 4) {
      v2f bf = *(const v2f*)(Bb + s);
      v2f a0 = *(const v2f*)(A0 + s);
      v2f a1 = *(const v2f*)(A1 + s);
      acc0 = __builtin_amdgcn_wmma_f32_16x16x4_f32(
          false, a0, false, bf, (short)0, acc0, false, false);
      acc1 = __builtin_amdgcn_wmma_f32_16x16x4_f32(
          false, a1, false, bf, (short)0, acc1, false, false);
    }

    __syncthreads();                          // all reads of this buffer done
    if (ch + 1 < NCHUNK) storeB((ch + 1) & 1);
    __syncthreads();                          // next buffer visible
  }

  // ------------------------ epilogue: truncate to i32 --------------------
  // C/D layout: VGPR v, lanes 0-15 -> M=v, lanes 16-31 -> M=v+8; N = lane&15
  const int ocol = nbase + (lane & 15);
  const int mhi  = (lane >> 4) * 8;
#pragma unroll
  for (int v = 0; v < 8; ++v) {
    int t0 = mbase0 + v + mhi;
    int t1 = mbase1 + v + mhi;
    out[(size_t)(t0 * C_OUT + ocol) * (ROWS * COLS) + rc] = (int)acc0[v];
    out[(size_t)(t1 * C_OUT + ocol) * (ROWS * COLS) + rc] = (int)acc1[v];
  }
}

extern "C" void kernel_launch(void* const* d_in, const int* in_sizes, int n_in,
                              void* d_out, int out_size, void* d_ws, size_t ws_size,
                              hipStream_t stream) {
  (void)in_sizes; (void)n_in; (void)out_size; (void)d_ws; (void)ws_size;
  const float* in  = (const float*)d_in[0];
  const float* wgt = (const float*)d_in[1];
  int* out = (int*)d_out;
  hipLaunchKernelGGL(localconv_wmma, dim3(ROWS * COLS), dim3(256), LDS_BYTES,
                     stream, in, wgt, out);
}